// NetBasic_57088705298790
// MI455X (gfx1250) — compile-verified
//
#include <hip/hip_runtime.h>
#include <math.h>

typedef float v2f __attribute__((ext_vector_type(2)));
typedef float v8f __attribute__((ext_vector_type(8)));

#define NFEAT   384
#define HID     284
#define HPAD    288
#define NCLS    6
#define NNODES  50000
#define NEDGES  250000
#define NGRAPHS 64
#define NT      6            // N-tiles (16 cols each) per wave; HPAD/(16*NT) = 3

// ---------------------------------------------------------------------------
// FP32 WMMA GEMM: C[M x HPAD] = A[M x K] * Bpack(K x HPAD)
// Bpack is fragment-major: Bpack[((k/4)*HPAD + n)*4 + (k&3)] = B[k][n],
// so a lane's 2-element B fragment is one contiguous 8B load.
// One wave per 16-row x 96-col tile. M%16==0, K%4==0, HPAD%96==0 -> no
// bounds checks, EXEC stays all ones (WMMA requirement).
// ---------------------------------------------------------------------------
template <int K>
__global__ __launch_bounds__(32) void gemm_wmma_f32(
    const float* __restrict__ A,          // [M x K], lda == K
    const float* __restrict__ Bpack,      // packed [K/4][HPAD][4]
    float* __restrict__ C)                // [M x HPAD]
{
    const int lane  = threadIdx.x;        // 0..31
    const int m0    = blockIdx.y * 16;
    const int n0    = blockIdx.x * (16 * NT);
    const int mrow  = lane & 15;          // M (A) / N (B,C) within tile
    const int khalf = (lane >> 4) * 2;    // lanes 0-15 -> K{0,1}; 16-31 -> K{2,3}

    v8f acc[NT];
    #pragma unroll
    for (int t = 0; t < NT; ++t) acc[t] = (v8f){0,0,0,0,0,0,0,0};

    const float* Arow = A + (size_t)(m0 + mrow) * K;

    #pragma unroll 2
    for (int k = 0; k < K; k += 4) {
        // A 16x4 fragment: lane holds A[m, k+khalf+{0,1}] (contiguous, 8B)
        const v2f a = *(const v2f*)(Arow + k + khalf);
        // B fragments: contiguous 8B per tile in packed layout
        const float* Bp = Bpack + ((size_t)(k >> 2) * HPAD + n0 + mrow) * 4 + khalf;
        #pragma unroll
        for (int t = 0; t < NT; ++t) {
            const v2f b = *(const v2f*)(Bp + t * 64);   // t*16 cols * 4
            acc[t] = __builtin_amdgcn_wmma_f32_16x16x4_f32(
                false, a, false, b, (short)0, acc[t], false, false);
        }
    }

    // C/D layout: VGPR v, lane l -> row m0 + v + 8*(l>>4), col n0 + t*16 + (l&15)
    const int mbase = m0 + 8 * (lane >> 4);
    #pragma unroll
    for (int t = 0; t < NT; ++t) {
        float* Cp = C + n0 + t * 16 + mrow;
        #pragma unroll
        for (int v = 0; v < 8; ++v)
            Cp[(size_t)(mbase + v) * HPAD] = acc[t][v];
    }
}

// ---------------------------------------------------------------------------
// Pack weights into fragment-major layout with zero padding:
//   dst[((k/4)*HPAD + n)*4 + (k&3)] = (k<rs && n<cs) ? W[k*cs+n] : 0
// kd = padded K (multiple of 4), covers n in [0, HPAD)
// ---------------------------------------------------------------------------
__global__ void pack_w(float* __restrict__ dst, const float* __restrict__ src,
                       int kd, int rs, int cs)
{
    int idx = blockIdx.x * blockDim.x + threadIdx.x;
    if (idx >= kd * HPAD) return;
    int k = idx / HPAD, n = idx - k * HPAD;
    float v = (k < rs && n < cs) ? src[k * cs + n] : 0.0f;
    dst[((size_t)(k >> 2) * HPAD + n) * 4 + (k & 3)] = v;
}

__global__ void pad_vec(float* __restrict__ dst, const float* __restrict__ src, int cs)
{
    int i = blockIdx.x * blockDim.x + threadIdx.x;
    if (i >= HPAD) return;
    dst[i] = (i < cs) ? src[i] : 0.0f;
}

// ---------------------------------------------------------------------------
// Degree accumulation (ns/nd pre-zeroed) and norm transform
// ---------------------------------------------------------------------------
__global__ void deg_kernel(const int* __restrict__ src, const int* __restrict__ dst,
                           float* __restrict__ deg_out, float* __restrict__ deg_in)
{
    int e = blockIdx.x * blockDim.x + threadIdx.x;
    if (e >= NEDGES) return;
    atomicAdd(&deg_out[src[e]], 1.0f);
    atomicAdd(&deg_in[dst[e]], 1.0f);
}

__global__ void norm_kernel(float* __restrict__ ns, float* __restrict__ nd)
{
    int i = blockIdx.x * blockDim.x + threadIdx.x;
    if (i >= NNODES) return;
    float a = ns[i];
    float b = nd[i];
    ns[i] = (a > 0.0f) ? rsqrtf(fmaxf(a, 1.0f)) : 0.0f;
    nd[i] = (b > 0.0f) ? rsqrtf(fmaxf(b, 1.0f)) : 0.0f;
}

// ---------------------------------------------------------------------------
// Edge scatter: agg[dst] += tmp[src] * ns[src]   (4 cols per thread)
// ---------------------------------------------------------------------------
__global__ void scatter_kernel(const float* __restrict__ tmp,
                               const float* __restrict__ ns,
                               const int* __restrict__ src,
                               const int* __restrict__ dst,
                               float* __restrict__ agg)
{
    const int CHUNKS = HPAD / 4;   // 72
    long long idx = (long long)blockIdx.x * blockDim.x + threadIdx.x;
    if (idx >= (long long)NEDGES * CHUNKS) return;
    int e = (int)(idx / CHUNKS);
    int j = (int)(idx - (long long)e * CHUNKS) * 4;
    int s = src[e], d = dst[e];
    float w = ns[s];
    const float4 v = *(const float4*)(tmp + (size_t)s * HPAD + j);
    float* ap = agg + (size_t)d * HPAD + j;
    atomicAdd(ap + 0, v.x * w);
    atomicAdd(ap + 1, v.y * w);
    atomicAdd(ap + 2, v.z * w);
    atomicAdd(ap + 3, v.w * w);
}

// ---------------------------------------------------------------------------
// h = relu(agg * nd + b)  (in place on agg)
// ---------------------------------------------------------------------------
__global__ void finalize_kernel(float* __restrict__ h,
                                const float* __restrict__ nd,
                                const float* __restrict__ b)
{
    long long idx = (long long)blockIdx.x * blockDim.x + threadIdx.x;
    if (idx >= (long long)NNODES * HPAD) return;
    int i = (int)(idx / HPAD);
    int c = (int)(idx - (long long)i * HPAD);
    float v = h[idx] * nd[i] + b[c];
    h[idx] = fmaxf(v, 0.0f);
}

// ---------------------------------------------------------------------------
// Per-graph pooling (sums) and counts
// ---------------------------------------------------------------------------
__global__ void pool_kernel(const float* __restrict__ h,
                            const int* __restrict__ gid,
                            float* __restrict__ gsum)
{
    const int CHUNKS = HPAD / 4;
    long long idx = (long long)blockIdx.x * blockDim.x + threadIdx.x;
    if (idx >= (long long)NNODES * CHUNKS) return;
    int i = (int)(idx / CHUNKS);
    int j = (int)(idx - (long long)i * CHUNKS) * 4;
    int g = gid[i];
    const float4 v = *(const float4*)(h + (size_t)i * HPAD + j);
    float* sp = gsum + (size_t)g * HPAD + j;
    atomicAdd(sp + 0, v.x);
    atomicAdd(sp + 1, v.y);
    atomicAdd(sp + 2, v.z);
    atomicAdd(sp + 3, v.w);
}

__global__ void count_kernel(const int* __restrict__ gid, float* __restrict__ gcnt)
{
    int i = blockIdx.x * blockDim.x + threadIdx.x;
    if (i >= NNODES) return;
    atomicAdd(&gcnt[gid[i]], 1.0f);
}

// ---------------------------------------------------------------------------
// Classifier: logits = [mean(h1) ; mean(h2)] @ Wc + bc, then log_softmax
// ---------------------------------------------------------------------------
__global__ void classifier_kernel(const float* __restrict__ gsum1, const float* __restrict__ gcnt1,
                                  const float* __restrict__ gsum2, const float* __restrict__ gcnt2,
                                  const float* __restrict__ Wc,    const float* __restrict__ bc,
                                  float* __restrict__ out)
{
    int g = threadIdx.x;
    if (g >= NGRAPHS) return;
    float inv1 = 1.0f / fmaxf(gcnt1[g], 1.0f);
    float inv2 = 1.0f / fmaxf(gcnt2[g], 1.0f);
    float logits[NCLS];
    #pragma unroll
    for (int c = 0; c < NCLS; ++c) logits[c] = bc[c];
    for (int k = 0; k < HID; ++k) {
        float h1 = gsum1[(size_t)g * HPAD + k] * inv1;
        #pragma unroll
        for (int c = 0; c < NCLS; ++c) logits[c] += h1 * Wc[k * NCLS + c];
    }
    for (int k = 0; k < HID; ++k) {
        float h2 = gsum2[(size_t)g * HPAD + k] * inv2;
        #pragma unroll
        for (int c = 0; c < NCLS; ++c) logits[c] += h2 * Wc[(HID + k) * NCLS + c];
    }
    float mx = logits[0];
    #pragma unroll
    for (int c = 1; c < NCLS; ++c) mx = fmaxf(mx, logits[c]);
    float se = 0.0f;
    #pragma unroll
    for (int c = 0; c < NCLS; ++c) se += expf(logits[c] - mx);
    float lse = mx + logf(se);
    #pragma unroll
    for (int c = 0; c < NCLS; ++c) out[g * NCLS + c] = logits[c] - lse;
}

// ---------------------------------------------------------------------------
extern "C" void kernel_launch(void* const* d_in, const int* in_sizes, int n_in,
                              void* d_out, int out_size, void* d_ws, size_t ws_size,
                              hipStream_t stream)
{
    const float* x1  = (const float*)d_in[0];
    const float* x2  = (const float*)d_in[1];
    const int* src1  = (const int*)d_in[2];
    const int* dst1  = (const int*)d_in[3];
    const int* src2  = (const int*)d_in[4];
    const int* dst2  = (const int*)d_in[5];
    const int* gid1  = (const int*)d_in[6];
    const int* gid2  = (const int*)d_in[7];
    // d_in[8] = num_graphs scalar (== 64, compile-time constant here)
    const float* W1  = (const float*)d_in[9];
    const float* b1  = (const float*)d_in[10];
    const float* W2  = (const float*)d_in[11];
    const float* b2  = (const float*)d_in[12];
    const float* W3  = (const float*)d_in[13];
    const float* b3  = (const float*)d_in[14];
    const float* Wc  = (const float*)d_in[15];
    const float* bc  = (const float*)d_in[16];

    float* ws = (float*)d_ws;
    size_t off = 0;
    float* B0    = ws + off; off += (size_t)NNODES * HPAD;   // GEMM output (tmp)
    float* B1    = ws + off; off += (size_t)NNODES * HPAD;   // agg / h
    float* W1p   = ws + off; off += (size_t)NFEAT * HPAD;    // packed
    float* W2p   = ws + off; off += (size_t)HPAD * HPAD;     // packed
    float* W3p   = ws + off; off += (size_t)HPAD * HPAD;     // packed
    float* b1p   = ws + off; off += HPAD;
    float* b2p   = ws + off; off += HPAD;
    float* b3p   = ws + off; off += HPAD;
    float* nsb   = ws + off; off += NNODES;
    float* ndb   = ws + off; off += NNODES;
    float* gsum1 = ws + off; off += (size_t)NGRAPHS * HPAD;
    float* gsum2 = ws + off; off += (size_t)NGRAPHS * HPAD;
    float* gcnt1 = ws + off; off += NGRAPHS;
    float* gcnt2 = ws + off; off += NGRAPHS;

    const int TPB = 256;
    // ---- pack fragment-major padded weights / biases ----
    pack_w<<<(NFEAT * HPAD + TPB - 1) / TPB, TPB, 0, stream>>>(W1p, W1, NFEAT, NFEAT, HID);
    pack_w<<<(HPAD * HPAD + TPB - 1) / TPB, TPB, 0, stream>>>(W2p, W2, HPAD, HID, HID);
    pack_w<<<(HPAD * HPAD + TPB - 1) / TPB, TPB, 0, stream>>>(W3p, W3, HPAD, HID, HID);
    pad_vec<<<(HPAD + TPB - 1) / TPB, TPB, 0, stream>>>(b1p, b1, HID);
    pad_vec<<<(HPAD + TPB - 1) / TPB, TPB, 0, stream>>>(b2p, b2, HID);
    pad_vec<<<(HPAD + TPB - 1) / TPB, TPB, 0, stream>>>(b3p, b3, HID);

    const dim3 gemm_grid(HPAD / (16 * NT), NNODES / 16);   // (3, 3125)
    const long long edge_work = (long long)NEDGES * (HPAD / 4);
    const long long node_work = (long long)NNODES * (HPAD / 4);
    const long long elem_work = (long long)NNODES * HPAD;

    for (int g = 0; g < 2; ++g) {
        const float* x   = (g == 0) ? x1   : x2;
        const int*   src = (g == 0) ? src1 : src2;
        const int*   dst = (g == 0) ? dst1 : dst2;
        const int*   gid = (g == 0) ? gid1 : gid2;
        float* gsum = (g == 0) ? gsum1 : gsum2;
        float* gcnt = (g == 0) ? gcnt1 : gcnt2;

        // ---- degrees -> norms ----
        hipMemsetAsync(nsb, 0, NNODES * sizeof(float), stream);
        hipMemsetAsync(ndb, 0, NNODES * sizeof(float), stream);
        deg_kernel<<<(NEDGES + TPB - 1) / TPB, TPB, 0, stream>>>(src, dst, nsb, ndb);
        norm_kernel<<<(NNODES + TPB - 1) / TPB, TPB, 0, stream>>>(nsb, ndb);

        // ---- 3 GraphConv layers ----
        for (int layer = 0; layer < 3; ++layer) {
            const float* bp = (layer == 0) ? b1p : (layer == 1) ? b2p : b3p;

            if (layer == 0)
                gemm_wmma_f32<NFEAT><<<gemm_grid, 32, 0, stream>>>(x, W1p, B0);
            else if (layer == 1)
                gemm_wmma_f32<HPAD><<<gemm_grid, 32, 0, stream>>>(B1, W2p, B0);
            else
                gemm_wmma_f32<HPAD><<<gemm_grid, 32, 0, stream>>>(B1, W3p, B0);

            hipMemsetAsync(B1, 0, (size_t)NNODES * HPAD * sizeof(float), stream);
            scatter_kernel<<<(int)((edge_work + TPB - 1) / TPB), TPB, 0, stream>>>(
                B0, nsb, src, dst, B1);
            finalize_kernel<<<(int)((elem_work + TPB - 1) / TPB), TPB, 0, stream>>>(B1, ndb, bp);
        }

        // ---- pooling ----
        hipMemsetAsync(gsum, 0, (size_t)NGRAPHS * HPAD * sizeof(float), stream);
        hipMemsetAsync(gcnt, 0, NGRAPHS * sizeof(float), stream);
        pool_kernel<<<(int)((node_work + TPB - 1) / TPB), TPB, 0, stream>>>(B1, gid, gsum);
        count_kernel<<<(NNODES + TPB - 1) / TPB, TPB, 0, stream>>>(gid, gcnt);
    }

    classifier_kernel<<<1, NGRAPHS, 0, stream>>>(gsum1, gcnt1, gsum2, gcnt2, Wc, bc,
                                                 (float*)d_out);
}